// HomogeneousGATLayer_61280593379541
// MI455X (gfx1250) — compile-verified
//
#include <hip/hip_runtime.h>
#include <math.h>

#define N_NODES 50000
#define N_EDGES 800000
#define IN_DIM  128
#define OUT_DIM 128
#define HEADS   4
#define D_K     32
#define ATT_DIM 65   // 2*D_K + 1

typedef __attribute__((ext_vector_type(2))) float v2f;
typedef __attribute__((ext_vector_type(8))) float v8f;

// ---------------------------------------------------------------------------
// Kernel 1: z = h @ W^T + b  using V_WMMA_F32_16X16X4_F32.
// One wave handles a 16-node strip and iterates over the 8 output 16-col tiles.
// A (16x4 f32) lane layout: lane l: m=l&15, g=l>>4 ; VGPR0=A[m][k0+2g], VGPR1=A[m][k0+2g+1]
// B (4x16 f32) lane layout (mirror): lane l: n=l&15, g=l>>4 ; VGPR0=B[k0+2g][n], VGPR1=B[k0+2g+1][n]
// C/D (16x16 f32): lane l, reg r: row = 8*(l>>4)+r, col = l&15
// ---------------------------------------------------------------------------
__global__ void __launch_bounds__(256)
gemm_z_wmma(const float* __restrict__ h, const float* __restrict__ W,
            const float* __restrict__ bias, float* __restrict__ z) {
    const int wave = (blockIdx.x * blockDim.x + threadIdx.x) >> 5;
    const int lane = threadIdx.x & 31;
    if (wave >= (N_NODES / 16)) return;   // whole-wave uniform exit -> EXEC all-1 for WMMA

    const int m = lane & 15;
    const int g = lane >> 4;

    // Preload the A strip for this lane: 32 k-steps, float2 each (contiguous pair).
    const float* __restrict__ hrow = h + (size_t)(wave * 16 + m) * IN_DIM;
    v2f a[32];
#pragma unroll
    for (int ks = 0; ks < 32; ++ks) {
        const int k = 4 * ks + 2 * g;
        a[ks].x = hrow[k];
        a[ks].y = hrow[k + 1];
    }

    for (int ot = 0; ot < 8; ++ot) {
        const int col = ot * 16 + m;                      // output column for B lane mapping
        const float* __restrict__ wrow = W + (size_t)col * IN_DIM;  // W[col][k] == B[k][col]
        v8f c = {};
#pragma unroll
        for (int ks = 0; ks < 32; ++ks) {
            const int k = 4 * ks + 2 * g;
            v2f b;
            b.x = wrow[k];
            b.y = wrow[k + 1];
            c = __builtin_amdgcn_wmma_f32_16x16x4_f32(
                    /*neg_a=*/false, a[ks], /*neg_b=*/false, b,
                    /*c_mod=*/(short)0, c, /*reuse_a=*/false, /*reuse_b=*/false);
        }
        const float bv = bias[col];
        const int rowBase = wave * 16 + g * 8;
#pragma unroll
        for (int r = 0; r < 8; ++r) {
            z[(size_t)(rowBase + r) * OUT_DIM + col] = c[r] + bv;
        }
    }
}

// ---------------------------------------------------------------------------
// Kernel 2: per-(node, head) attention projections
//   s_src[n,h] = dot(z[n, h*32 : h*32+32], att[h, 0:32])
//   s_dst[n,h] = dot(z[n, h*32 : h*32+32], att[h, 32:64])
// ---------------------------------------------------------------------------
__global__ void __launch_bounds__(256)
node_scores(const float* __restrict__ z, const float* __restrict__ att,
            float* __restrict__ s_src, float* __restrict__ s_dst) {
    const int gid = blockIdx.x * blockDim.x + threadIdx.x;
    if (gid >= N_NODES * HEADS) return;
    const int hd = gid & 3;
    const int n  = gid >> 2;
    const float* __restrict__ zp = z + (size_t)n * OUT_DIM + hd * D_K;
    const float* __restrict__ as = att + hd * ATT_DIM;
    const float* __restrict__ ad = as + D_K;
    float ss = 0.f, sd = 0.f;
#pragma unroll
    for (int k = 0; k < D_K; ++k) {
        const float v = zp[k];
        ss = fmaf(v, as[k], ss);
        sd = fmaf(v, ad[k], sd);
    }
    s_src[gid] = ss;
    s_dst[gid] = sd;
}

// ---------------------------------------------------------------------------
// Kernel 3: zero a float range (denom + accum are contiguous in ws)
// ---------------------------------------------------------------------------
__global__ void __launch_bounds__(256)
zero_f32(float* __restrict__ p, int n) {
    const int gid = blockIdx.x * blockDim.x + threadIdx.x;
    if (gid < n) p[gid] = 0.f;
}

__device__ __forceinline__ float leaky_clip_exp(float s) {
    s = (s >= 0.f) ? s : 0.2f * s;
    s = fminf(fmaxf(s, -20.f), 20.f);
    return __expf(s);
}

// ---------------------------------------------------------------------------
// Kernel 4: per-(edge, head) attention numerator + denominator segment-sum
// ---------------------------------------------------------------------------
__global__ void __launch_bounds__(256)
edge_denom(const int* __restrict__ ei, const float* __restrict__ ef,
           const float* __restrict__ s_src, const float* __restrict__ s_dst,
           const float* __restrict__ att, float* __restrict__ denom) {
    const int gid = blockIdx.x * blockDim.x + threadIdx.x;
    if (gid >= N_EDGES * HEADS) return;
    const int hd = gid & 3;
    const int e  = gid >> 2;
    const int src = ei[e];
    const int dst = ei[N_EDGES + e];
    const float s = s_src[src * HEADS + hd] + s_dst[dst * HEADS + hd]
                  + ef[e] * att[hd * ATT_DIM + 2 * D_K];
    const float alpha = leaky_clip_exp(s);
    __hip_atomic_fetch_add(&denom[dst * HEADS + hd], alpha,
                           __ATOMIC_RELAXED, __HIP_MEMORY_SCOPE_AGENT);
}

// ---------------------------------------------------------------------------
// Kernel 5: wave-per-edge aggregation: accum[dst] += z[src] * norm_alpha
// Lane l owns channels 4l..4l+3 (all in head l>>3); 4 fp32 atomics per lane.
// ---------------------------------------------------------------------------
__global__ void __launch_bounds__(256)
edge_aggregate(const int* __restrict__ ei, const float* __restrict__ ef,
               const float* __restrict__ s_src, const float* __restrict__ s_dst,
               const float* __restrict__ att, const float* __restrict__ denom,
               const float* __restrict__ z, float* __restrict__ accum) {
    const int wid  = (blockIdx.x * blockDim.x + threadIdx.x) >> 5;
    const int lane = threadIdx.x & 31;
    if (wid >= N_EDGES) return;
    const int src = ei[wid];
    const int dst = ei[N_EDGES + wid];
    const int hd  = lane >> 3;   // head for channels 4*lane .. 4*lane+3

    const float s = s_src[src * HEADS + hd] + s_dst[dst * HEADS + hd]
                  + ef[wid] * att[hd * ATT_DIM + 2 * D_K];
    const float alpha = leaky_clip_exp(s);
    const float na = alpha / (denom[dst * HEADS + hd] + 1e-6f);

    const float4 zv = *(const float4*)(z + (size_t)src * OUT_DIM + lane * 4);
    float* __restrict__ ap = accum + (size_t)dst * OUT_DIM + lane * 4;
    __hip_atomic_fetch_add(ap + 0, zv.x * na, __ATOMIC_RELAXED, __HIP_MEMORY_SCOPE_AGENT);
    __hip_atomic_fetch_add(ap + 1, zv.y * na, __ATOMIC_RELAXED, __HIP_MEMORY_SCOPE_AGENT);
    __hip_atomic_fetch_add(ap + 2, zv.z * na, __ATOMIC_RELAXED, __HIP_MEMORY_SCOPE_AGENT);
    __hip_atomic_fetch_add(ap + 3, zv.w * na, __ATOMIC_RELAXED, __HIP_MEMORY_SCOPE_AGENT);
}

// ---------------------------------------------------------------------------
// Kernel 6: wave-per-node residual + LayerNorm + ELU
// ---------------------------------------------------------------------------
__global__ void __launch_bounds__(256)
finalize(const float* __restrict__ accum, const float* __restrict__ z,
         const float* __restrict__ ln_g, const float* __restrict__ ln_b,
         float* __restrict__ out) {
    const int wid  = (blockIdx.x * blockDim.x + threadIdx.x) >> 5;
    const int lane = threadIdx.x & 31;
    if (wid >= N_NODES) return;

    const size_t base = (size_t)wid * OUT_DIM + lane * 4;
    const float4 a  = *(const float4*)(accum + base);
    const float4 zv = *(const float4*)(z + base);
    float x[4] = { a.x + zv.x, a.y + zv.y, a.z + zv.z, a.w + zv.w };

    float s = x[0] + x[1] + x[2] + x[3];
#pragma unroll
    for (int off = 16; off > 0; off >>= 1) s += __shfl_xor(s, off, 32);
    const float mu = s * (1.f / 128.f);

    float v = 0.f;
#pragma unroll
    for (int i = 0; i < 4; ++i) { const float d = x[i] - mu; v = fmaf(d, d, v); }
#pragma unroll
    for (int off = 16; off > 0; off >>= 1) v += __shfl_xor(v, off, 32);
    const float rstd = rsqrtf(v * (1.f / 128.f) + 1e-5f);

    const float4 gv = *(const float4*)(ln_g + lane * 4);
    const float4 bv = *(const float4*)(ln_b + lane * 4);
    const float g4[4] = { gv.x, gv.y, gv.z, gv.w };
    const float b4[4] = { bv.x, bv.y, bv.z, bv.w };

    float y[4];
#pragma unroll
    for (int i = 0; i < 4; ++i) {
        const float t = (x[i] - mu) * rstd * g4[i] + b4[i];
        y[i] = (t > 0.f) ? t : (__expf(t) - 1.f);   // ELU(alpha=1)
    }
    float4 o = { y[0], y[1], y[2], y[3] };
    *(float4*)(out + base) = o;
}

// ---------------------------------------------------------------------------
extern "C" void kernel_launch(void* const* d_in, const int* in_sizes, int n_in,
                              void* d_out, int out_size, void* d_ws, size_t ws_size,
                              hipStream_t stream) {
    const float* h    = (const float*)d_in[0];   // [50000,128]
    const int*   ei   = (const int*)  d_in[1];   // [2,800000] (int32: JAX x64 disabled)
    const float* ef   = (const float*)d_in[2];   // [800000,1]
    const float* W_w  = (const float*)d_in[3];   // [128,128]
    const float* W_b  = (const float*)d_in[4];   // [128]
    const float* att  = (const float*)d_in[5];   // [4,65]
    const float* ln_g = (const float*)d_in[6];   // [128]
    const float* ln_b = (const float*)d_in[7];   // [128]
    float* out = (float*)d_out;                  // [50000,128]

    // Workspace partition (floats): z | s_src | s_dst | denom | accum
    float* ws     = (float*)d_ws;
    float* z      = ws;                                    // 6,400,000
    float* s_src  = z      + (size_t)N_NODES * OUT_DIM;    //   200,000
    float* s_dst  = s_src  + (size_t)N_NODES * HEADS;      //   200,000
    float* denom  = s_dst  + (size_t)N_NODES * HEADS;      //   200,000
    float* accum  = denom  + (size_t)N_NODES * HEADS;      // 6,400,000

    // 1) GEMM via fp32 WMMA: 3125 waves, 8 waves/block
    {
        const int waves = N_NODES / 16;                    // 3125
        const int blocks = (waves + 7) / 8;
        gemm_z_wmma<<<blocks, 256, 0, stream>>>(h, W_w, W_b, z);
    }
    // 2) per-(node,head) projections
    node_scores<<<(N_NODES * HEADS + 255) / 256, 256, 0, stream>>>(z, att, s_src, s_dst);
    // 3) zero denom+accum (contiguous range)
    {
        const int nz = N_NODES * HEADS + N_NODES * OUT_DIM;  // 6,600,000
        zero_f32<<<(nz + 255) / 256, 256, 0, stream>>>(denom, nz);
    }
    // 4) attention denominators
    edge_denom<<<(N_EDGES * HEADS + 255) / 256, 256, 0, stream>>>(ei, ef, s_src, s_dst, att, denom);
    // 5) weighted aggregation (wave per edge)
    edge_aggregate<<<(N_EDGES * 32 + 255) / 256, 256, 0, stream>>>(ei, ef, s_src, s_dst, att,
                                                                   denom, z, accum);
    // 6) residual + LayerNorm + ELU (wave per node)
    finalize<<<(N_NODES * 32 + 255) / 256, 256, 0, stream>>>(accum, z, ln_g, ln_b, out);
}